// MPNN_Layer_46076409151747
// MI455X (gfx1250) — compile-verified
//
#include <hip/hip_runtime.h>

typedef __attribute__((ext_vector_type(2))) float v2f;
typedef __attribute__((ext_vector_type(8))) float v8f;

#define D_FEAT 64

// ---------------------------------------------------------------------------
// Kernel 1: zero the ft accumulator (d_ws) with wide stores.
// ---------------------------------------------------------------------------
__global__ void mpnn_zero_ft(float* __restrict__ ft, int n_floats) {
    int i = (blockIdx.x * blockDim.x + threadIdx.x) * 4;
    if (i < n_floats) {
        *(float4*)(ft + i) = make_float4(0.f, 0.f, 0.f, 0.f);
    }
}

// ---------------------------------------------------------------------------
// Kernel 2: edge scatter.  One wave32 per edge.
//   m = h[src] * e  ;  ft[dst] += m   (global_atomic_add_f32, L2-resident)
// Each lane covers 2 features: coalesced float2 gather, 2 f32 atomics.
// ---------------------------------------------------------------------------
__global__ void mpnn_edge_scatter(const float* __restrict__ h,
                                  const float* __restrict__ e,
                                  const int*   __restrict__ src,
                                  const int*   __restrict__ dst,
                                  float*       __restrict__ ft,
                                  int n_edges) {
    int wave = (blockIdx.x * blockDim.x + threadIdx.x) >> 5;
    int lane = threadIdx.x & 31;
    if (wave >= n_edges) return;

    int   s  = src[wave];          // wave-uniform loads (broadcast from cache)
    int   d  = dst[wave];
    float ev = e[wave];

    const float2 hv = ((const float2*)(h + (long long)s * D_FEAT))[lane];
    float* outp = ft + (long long)d * D_FEAT + lane * 2;
    atomicAdd(outp,     hv.x * ev);
    atomicAdd(outp + 1, hv.y * ev);
}

// ---------------------------------------------------------------------------
// Kernel 3: out = ft @ W^T + b via V_WMMA_F32_16X16X4_F32.
// One wave computes one 16x16 output tile; K=64 unrolled as 16 WMMA steps.
//
// 32-bit A (16x4, MxK) layout: lanes 0-15 = rows, VGPR0/1 = K{0,1} (half 0)
// or K{2,3} (half 1)  -> contiguous float2 load from ft row.
// 32-bit B (4x16, KxN) layout mirrors A with N across lanes; B[k][n] = W[n][k]
// -> contiguous float2 load from W row n.
// C/D layout: VGPR r holds row (r + 8*half), column = lane&15.
// Bias folded in by seeding the accumulator with b[n].
// ---------------------------------------------------------------------------
__global__ void mpnn_linear_wmma(const float* __restrict__ ft,
                                 const float* __restrict__ W,
                                 const float* __restrict__ b,
                                 float*       __restrict__ out,
                                 int n_mtiles) {
    int gwave = (blockIdx.x * blockDim.x + threadIdx.x) >> 5;
    int lane  = threadIdx.x & 31;
    int ntile = gwave & 3;        // 4 tiles of 16 output features
    int mtile = gwave >> 2;       // tiles of 16 nodes
    if (mtile >= n_mtiles) return;   // wave-uniform; never splits EXEC

    const int row0 = mtile * 16;
    const int n0   = ntile * 16;
    const int half = lane >> 4;
    const int l16  = lane & 15;

    const float bias = b[n0 + l16];
    v8f c;
#pragma unroll
    for (int r = 0; r < 8; ++r) c[r] = bias;

    const float* arow = ft + (long long)(row0 + l16) * D_FEAT + 2 * half;
    const float* brow = W  + (long long)(n0   + l16) * D_FEAT + 2 * half;

#pragma unroll
    for (int k0 = 0; k0 < D_FEAT; k0 += 4) {
        v2f a  = *(const v2f*)(arow + k0);
        v2f bb = *(const v2f*)(brow + k0);
        // 8 args: (neg_a, A, neg_b, B, c_mod, C, reuse_a, reuse_b)
        c = __builtin_amdgcn_wmma_f32_16x16x4_f32(
                false, a, false, bb, (short)0, c, false, false);
    }

    float* orow = out + (long long)(row0 + 8 * half) * D_FEAT + n0 + l16;
#pragma unroll
    for (int r = 0; r < 8; ++r) orow[(long long)r * D_FEAT] = c[r];
}

// ---------------------------------------------------------------------------
// Launcher
// ---------------------------------------------------------------------------
extern "C" void kernel_launch(void* const* d_in, const int* in_sizes, int n_in,
                              void* d_out, int out_size, void* d_ws, size_t ws_size,
                              hipStream_t stream) {
    const float* h   = (const float*)d_in[0];
    const float* e   = (const float*)d_in[1];
    const int*   src = (const int*)  d_in[2];
    const int*   dst = (const int*)  d_in[3];
    const float* W   = (const float*)d_in[4];
    const float* b   = (const float*)d_in[5];
    float*       out = (float*)d_out;
    float*       ft  = (float*)d_ws;           // [N_NODES, 64] f32 accumulator

    const int n_edges  = in_sizes[1];          // e has one value per edge
    const int n_nodes  = in_sizes[0] / D_FEAT; // h is [N, 64]
    const int ft_elems = n_nodes * D_FEAT;

    // 1) ft = 0
    {
        int threads = (ft_elems + 3) / 4;
        int blocks  = (threads + 255) / 256;
        mpnn_zero_ft<<<blocks, 256, 0, stream>>>(ft, ft_elems);
    }

    // 2) ft[dst] += h[src] * e   (one wave per edge)
    {
        long long total_threads = (long long)n_edges * 32;
        int blocks = (int)((total_threads + 255) / 256);
        mpnn_edge_scatter<<<blocks, 256, 0, stream>>>(h, e, src, dst, ft, n_edges);
    }

    // 3) out = ft @ W^T + b   (one wave per 16x16 tile)
    {
        int n_mtiles = (n_nodes + 15) / 16;        // 6250 for N=100000
        long long total_threads = (long long)n_mtiles * 4 * 32;
        int blocks = (int)((total_threads + 255) / 256);
        mpnn_linear_wmma<<<blocks, 256, 0, stream>>>(ft, W, b, out, n_mtiles);
    }
}